// CustomConvolve_46583215292444
// MI455X (gfx1250) — compile-verified
//
#include <hip/hip_runtime.h>
#include <cstdint>

// Problem constants from the reference: x [16,16,512,512] f32,
// weights [(511*511),4] f32, bias [511*511] f32, out [16,16,511,511] f32.
#define NBC   256        // B*C
#define WDIM  512
#define HDIM  512
#define ODIM  511        // output W-1 == H-1
#define TW    16         // output tile rows (w)
#define TH    64         // output tile cols (h)
#define ROWS  (TW + 1)   // x tile rows staged in LDS
#define SCOLS 68         // staged cols: 17 float4s per row, 16B aligned
#define SQ4   17         // float4s per staged row
#define NQ4   (ROWS * SQ4)  // 289 float4 elements per tile

__global__ __launch_bounds__(256)
void lc_conv2x2_async(const float* __restrict__ x,
                      const float* __restrict__ weights,
                      const float* __restrict__ bias,
                      float* __restrict__ out)
{
    __shared__ float tile[ROWS * SCOLS];   // 17*68*4 = 4624 B

    const int tid = threadIdx.x;
    const int oh0 = blockIdx.x * TH;   // output h tile origin
    const int ow0 = blockIdx.y * TW;   // output w tile origin
    const int bc  = blockIdx.z;        // fused batch*channel index

    // Staged column window: [cstart, cstart+67]. Both branches are multiples
    // of 4 floats -> 16B-aligned global rows; window never exceeds col 511
    // (max oh0 = 448 -> cstart = 444 -> last col = 511).
    const int cstart = (oh0 == 0) ? 0 : oh0 - 4;
    const int cofs   = (oh0 == 0) ? -1 : 3;   // tile col of global col (oh-1)

    const float* xbc = x + (size_t)bc * (WDIM * HDIM);

    // --- Stage ROWS x SCOLS input window into LDS via CDNA5 async B128 DMA ---
    // tile[r][4*c4 .. 4*c4+3] <- x[bc][clamp(ow0-1+r)][cstart + 4*c4 ..]
#pragma unroll
    for (int i = 0; i < 2; ++i) {
        const int e = tid + i * 256;
        if (e < NQ4) {
            const int r  = e / SQ4;
            const int c4 = e - r * SQ4;
            int gw = ow0 - 1 + r;
            gw = gw < 0 ? 0 : gw;                 // only w-tile 0 clamps
            const float* gp   = xbc + gw * HDIM + cstart + c4 * 4;
            const uint32_t ldsb =
                (uint32_t)(uintptr_t)(&tile[r * SCOLS + c4 * 4]);
            // ASYNCcnt-tracked 16B-per-lane DMA into LDS (512B per wave).
            asm volatile("global_load_async_to_lds_b128 %0, %1, off"
                         :: "v"(ldsb), "v"(gp)
                         : "memory");
        }
    }
    asm volatile("s_wait_asynccnt 0" ::: "memory");
    __syncthreads();

    // --- Compute: each thread -> 4 outputs along w at one h ---
    const int lh = tid & (TH - 1);   // 0..63  local h
    const int wg = tid >> 6;         // 0..3   w group
    const int oh = oh0 + lh;

    // tile cols for global cols (oh-1, oh). c0 == -1 only for the zero-forced
    // border output (oh0==0, lh==0); clamp keeps the dead read in range.
    int c0 = lh + cofs;
    c0 = c0 < 0 ? 0 : c0;
    const int c1 = lh + cofs + 1;    // 0..67, always in range

    const float4* __restrict__ w4 = (const float4*)weights;
    float* __restrict__ obc = out + (size_t)bc * (ODIM * ODIM);

#pragma unroll
    for (int i = 0; i < 4; ++i) {
        const int lw = wg * 4 + i;
        const int ow = ow0 + lw;
        if (ow > ODIM - 1 || oh > ODIM - 1) continue;  // outside output

        float v = 0.0f;
        if (ow != 0 && oh != 0) {
            // patch (row-major): x[ow-1,oh-1], x[ow-1,oh], x[ow,oh-1], x[ow,oh]
            const float x00 = tile[ lw      * SCOLS + c0];
            const float x01 = tile[ lw      * SCOLS + c1];
            const float x10 = tile[(lw + 1) * SCOLS + c0];
            const float x11 = tile[(lw + 1) * SCOLS + c1];
            const int   idx = ODIM * ow + oh;        // neuron index
            const float4 wv = w4[idx];               // 16B aligned, L2-hot
            v = fmaf(x00, wv.x,
                fmaf(x01, wv.y,
                fmaf(x10, wv.z,
                     x11 * wv.w))) + bias[idx];
        }
        // Streaming write-once output: non-temporal so it doesn't rinse the
        // 256x-reused weights/bias and the x-tile overlap window out of L2.
        __builtin_nontemporal_store(v, &obc[(size_t)ow * ODIM + oh]);
    }
}

extern "C" void kernel_launch(void* const* d_in, const int* in_sizes, int n_in,
                              void* d_out, int out_size, void* d_ws, size_t ws_size,
                              hipStream_t stream)
{
    (void)in_sizes; (void)n_in; (void)out_size; (void)d_ws; (void)ws_size;

    const float* x       = (const float*)d_in[0];
    const float* weights = (const float*)d_in[1];
    const float* bias    = (const float*)d_in[2];
    float*       out     = (float*)d_out;

    // h tiles fastest, then w tiles, bc slowest -> maximal L2 reuse of
    // weights/bias and of x rows shared between adjacent w tiles.
    dim3 grid((ODIM + TH - 1) / TH,   // 8
              (ODIM + TW - 1) / TW,   // 32
              NBC);                   // 256
    lc_conv2x2_async<<<grid, dim3(256, 1, 1), 0, stream>>>(x, weights, bias, out);
}